// MultiGraphEncoder_37417755083213
// MI455X (gfx1250) — compile-verified
//
#include <hip/hip_runtime.h>

#define N_NODES 100000
#define HDIM    128
#define LDS_STRIDE 132   // padded stride (floats): 132%64==4 -> conflict-free column reads

typedef __attribute__((ext_vector_type(2))) float v2f;
typedef __attribute__((ext_vector_type(8))) float v8f;

__device__ __forceinline__ void atomAddF(float* p, float v) {
    // hardware global_atomic_add_f32 (no return -> STOREcnt path)
    unsafeAtomicAdd(p, v);
}

// ---------------- degree / inverse degree ----------------
__global__ void degree_kernel(const int* __restrict__ src, float* deg, int nE) {
    int i = blockIdx.x * blockDim.x + threadIdx.x;
    if (i < nE) atomAddF(&deg[src[i]], 1.0f);
}

__global__ void inv_kernel(float* d, int n) {
    int i = blockIdx.x * blockDim.x + threadIdx.x;
    if (i < n) d[i] = 1.0f / fmaxf(d[i], 1.0f);
}

// ---------------- 128x128 weight transpose: Wt[n][k] = W[k][n] ----------------
__global__ void transpose128_kernel(const float* __restrict__ in, float* __restrict__ outp) {
    int tid = blockIdx.x * blockDim.x + threadIdx.x;   // 64 blocks x 256 = 16384
    int k = tid >> 7, n = tid & 127;
    outp[n * HDIM + k] = in[k * HDIM + n];
}

// ---------------- SpMM: agg[dst] += z[src] * inv[src] ----------------
// one wave32 per edge; each lane owns a float4 (128 = 32*4)
__global__ void spmm_kernel(const float* __restrict__ z, const int* __restrict__ src,
                            const int* __restrict__ dst, const float* __restrict__ inv,
                            float* agg, int nE) {
    int lane = threadIdx.x & 31;
    int e = blockIdx.x * (blockDim.x >> 5) + (threadIdx.x >> 5);
    if (e >= nE) return;
    int s = src[e];
    int d = dst[e];
    float sc = inv[s];
    float4 v = *((const float4*)(z + (size_t)s * HDIM) + lane);
    float* ap = agg + (size_t)d * HDIM + lane * 4;
    atomAddF(ap + 0, v.x * sc);
    atomAddF(ap + 1, v.y * sc);
    atomAddF(ap + 2, v.z * sc);
    atomAddF(ap + 3, v.w * sc);
}

// ---------------- GEMM: out[N,128] = act(A[N,128] @ W[128,128] + b) ----------------
// Takes the PRE-TRANSPOSED weight Wt[n][k]. act: 0 = none, 1 = PReLU(slope), 2 = ELU.
// Block = 256 threads = 8 waves computing a 128x128 output tile. A tile and Wt are
// staged into LDS with GLOBAL_LOAD_ASYNC_TO_LDS_B128 (ASYNCcnt path), then each wave
// runs a K-loop of v_wmma_f32_16x16x4_f32 with single ds_load_b64 fragment loads.
__launch_bounds__(256)
__global__ void gemm128_kernel(const float* __restrict__ A, const float* __restrict__ Wt,
                               const float* __restrict__ bias, float* out,
                               int nRows, int act, const float* __restrict__ slope_ptr) {
    extern __shared__ float lds[];
    float* As  = lds;                        // 128 x LDS_STRIDE  (row-major A tile)
    float* Wtl = lds + 128 * LDS_STRIDE;     // 128 x LDS_STRIDE  (Wt: [n][k])

    const int tid = threadIdx.x;
    const int m0  = blockIdx.x * 128;

    // async-stage both tiles: 16B granules, per-lane LDS byte address in VDST.
    // Row stride 132 floats = 528 B (16B-aligned), so B128 granules stay aligned.
    for (int idx = tid; idx < 128 * 32; idx += 256) {
        int row = idx >> 5, c4 = (idx & 31) << 2;
        int gr = m0 + row; if (gr >= nRows) gr = nRows - 1;
        unsigned lA = (unsigned)(size_t)(const void*)&As[row * LDS_STRIDE + c4];
        unsigned long long gA = (unsigned long long)(const void*)(A + (size_t)gr * HDIM + c4);
        asm volatile("global_load_async_to_lds_b128 %0, %1, off"
                     :: "v"(lA), "v"(gA) : "memory");
        unsigned lW = (unsigned)(size_t)(const void*)&Wtl[row * LDS_STRIDE + c4];
        unsigned long long gW = (unsigned long long)(const void*)(Wt + (size_t)row * HDIM + c4);
        asm volatile("global_load_async_to_lds_b128 %0, %1, off"
                     :: "v"(lW), "v"(gW) : "memory");
    }
    asm volatile("s_wait_asynccnt 0" ::: "memory");
    __syncthreads();

    const int wave = tid >> 5;
    const int lane = tid & 31;
    const int mrow = lane & 15;             // M index (A frag) / N index (B frag)
    const int koff = (lane >> 4) << 1;      // 0 for lanes 0-15, 2 for lanes 16-31

    v8f acc[8];
#pragma unroll
    for (int nt = 0; nt < 8; ++nt) acc[nt] = (v8f){0,0,0,0,0,0,0,0};

    // A 16x4 frag: lane=M, VGPR0/1 = K{koff,koff+1} -> contiguous 8B load
    // B 4x16 frag: lane=N, VGPR0/1 = K{koff,koff+1} -> contiguous 8B load from Wt[n][k]
    const float* arow = &As[(wave * 16 + mrow) * LDS_STRIDE + koff];
    const float* wrow = &Wtl[mrow * LDS_STRIDE + koff];
    for (int k = 0; k < 128; k += 4) {
        v2f a = *(const v2f*)(arow + k);
#pragma unroll
        for (int nt = 0; nt < 8; ++nt) {
            v2f b = *(const v2f*)(wrow + nt * (16 * LDS_STRIDE) + k);
            acc[nt] = __builtin_amdgcn_wmma_f32_16x16x4_f32(
                false, a, false, b, (short)0, acc[nt], false, false);
        }
    }

    const float sl = slope_ptr[0];
    // C/D layout: VGPR r -> M = r (lanes 0-15) / r+8 (lanes 16-31), N = lane%16
    const int mbase = m0 + wave * 16 + ((lane >> 4) << 3);
#pragma unroll
    for (int nt = 0; nt < 8; ++nt) {
        int col = nt * 16 + mrow;
        float bv = bias[col];
#pragma unroll
        for (int r = 0; r < 8; ++r) {
            int m = mbase + r;
            if (m < nRows) {
                float v = acc[nt][r] + bv;
                if (act == 1)      v = (v >= 0.f) ? v : sl * v;
                else if (act == 2) v = (v >= 0.f) ? v : (__expf(v) - 1.f);
                out[(size_t)m * HDIM + col] = v;
            }
        }
    }
}

// ---------------- row L2 normalize (one wave per row) ----------------
__global__ void l2norm_kernel(const float* __restrict__ in, float* out, int n) {
    int lane = threadIdx.x & 31;
    int row = blockIdx.x * (blockDim.x >> 5) + (threadIdx.x >> 5);
    if (row >= n) return;
    float4 v = *((const float4*)(in + (size_t)row * HDIM) + lane);
    float s = v.x * v.x + v.y * v.y + v.z * v.z + v.w * v.w;
#pragma unroll
    for (int off = 16; off; off >>= 1) s += __shfl_xor(s, off, 32);
    float sc = 1.0f / fmaxf(sqrtf(s), 1e-12f);
    float4 o = { v.x * sc, v.y * sc, v.z * sc, v.w * sc };
    *((float4*)(out + (size_t)row * HDIM) + lane) = o;
}

// ---------------- cluster head tail: l2norm(softmax(h @ w2 + b2)), K = 3 ----------------
__global__ void cluster_kernel(const float* __restrict__ h, const float* __restrict__ w2,
                               const float* __restrict__ b2, float* out, int n) {
    int lane = threadIdx.x & 31;
    int row = blockIdx.x * (blockDim.x >> 5) + (threadIdx.x >> 5);
    if (row >= n) return;
    float4 v = *((const float4*)(h + (size_t)row * HDIM) + lane);
    int k0 = lane * 4;
    float p0 = v.x * w2[(k0+0)*3+0] + v.y * w2[(k0+1)*3+0] + v.z * w2[(k0+2)*3+0] + v.w * w2[(k0+3)*3+0];
    float p1 = v.x * w2[(k0+0)*3+1] + v.y * w2[(k0+1)*3+1] + v.z * w2[(k0+2)*3+1] + v.w * w2[(k0+3)*3+1];
    float p2 = v.x * w2[(k0+0)*3+2] + v.y * w2[(k0+1)*3+2] + v.z * w2[(k0+2)*3+2] + v.w * w2[(k0+3)*3+2];
#pragma unroll
    for (int off = 16; off; off >>= 1) {
        p0 += __shfl_xor(p0, off, 32);
        p1 += __shfl_xor(p1, off, 32);
        p2 += __shfl_xor(p2, off, 32);
    }
    if (lane == 0) {
        p0 += b2[0]; p1 += b2[1]; p2 += b2[2];
        float mx = fmaxf(p0, fmaxf(p1, p2));
        float e0 = __expf(p0 - mx), e1 = __expf(p1 - mx), e2 = __expf(p2 - mx);
        float is = 1.0f / (e0 + e1 + e2);
        e0 *= is; e1 *= is; e2 *= is;
        float sc = 1.0f / fmaxf(sqrtf(e0*e0 + e1*e1 + e2*e2), 1e-12f);
        float* o = out + (size_t)row * 3;
        o[0] = e0 * sc; o[1] = e1 * sc; o[2] = e2 * sc;
    }
}

extern "C" void kernel_launch(void* const* d_in, const int* in_sizes, int n_in,
                              void* d_out, int out_size, void* d_ws, size_t ws_size,
                              hipStream_t stream) {
    const float* x[3]  = { (const float*)d_in[0], (const float*)d_in[1], (const float*)d_in[2] };
    const int*   e[3]  = { (const int*)d_in[3],   (const int*)d_in[4],   (const int*)d_in[5]   };
    int nE[3] = { in_sizes[3] / 2, in_sizes[4] / 2, in_sizes[5] / 2 };
    const float* Ws   = (const float*)d_in[6];
    const float* bs   = (const float*)d_in[7];
    const float* a_p  = (const float*)d_in[8];
    const float* cw1  = (const float*)d_in[9];
    const float* cb1  = (const float*)d_in[10];
    const float* cw2  = (const float*)d_in[11];
    const float* cb2  = (const float*)d_in[12];
    const float* iw1  = (const float*)d_in[13];
    const float* ib1  = (const float*)d_in[14];
    const float* iw2  = (const float*)d_in[15];
    const float* ib2  = (const float*)d_in[16];
    float* out = (float*)d_out;

    const size_t NH = (size_t)N_NODES * HDIM;
    const size_t WSZ = (size_t)HDIM * HDIM;
    float* buf0 = (float*)d_ws;
    float* buf1 = buf0 + NH;
    float* invb = buf1 + NH;
    float* wt   = invb + N_NODES;     // 6 transposed 128x128 weights
    float* wtW[3] = { wt, wt + WSZ, wt + 2 * WSZ };
    float* wti1 = wt + 3 * WSZ;
    float* wti2 = wt + 4 * WSZ;
    float* wtc1 = wt + 5 * WSZ;

    // pre-transpose all GEMM weights (Wt[n][k]) so B fragments are contiguous b64 loads
    for (int l = 0; l < 3; ++l)
        transpose128_kernel<<<64, 256, 0, stream>>>(Ws + (size_t)l * WSZ, wtW[l]);
    transpose128_kernel<<<64, 256, 0, stream>>>(iw1, wti1);
    transpose128_kernel<<<64, 256, 0, stream>>>(iw2, wti2);
    transpose128_kernel<<<64, 256, 0, stream>>>(cw1, wtc1);

    const int gemmGrid = (N_NODES + 127) / 128;
    const int rowGrid  = (N_NODES + 7) / 8;          // 8 waves / 256-thread block
    const size_t gemmLds = (size_t)(2 * 128 * LDS_STRIDE) * sizeof(float);

    for (int g = 0; g < 3; ++g) {
        const int* src = e[g];
        const int* dst = e[g] + nE[g];

        hipMemsetAsync(invb, 0, N_NODES * sizeof(float), stream);
        degree_kernel<<<(nE[g] + 255) / 256, 256, 0, stream>>>(src, invb, nE[g]);
        inv_kernel<<<(N_NODES + 255) / 256, 256, 0, stream>>>(invb, N_NODES);

        const float* zin = x[g];
        float* zbufs[3] = { buf0, buf1, buf0 };
        for (int l = 0; l < 3; ++l) {
            float* agg = zbufs[l];
            hipMemsetAsync(agg, 0, NH * sizeof(float), stream);
            spmm_kernel<<<(nE[g] + 7) / 8, 256, 0, stream>>>(zin, src, dst, invb, agg, nE[g]);
            // in-place GEMM is safe: each block stages its rows in LDS before writing
            gemm128_kernel<<<gemmGrid, 256, gemmLds, stream>>>(
                agg, wtW[l], bs + (size_t)l * HDIM, agg, N_NODES, /*PReLU*/1, a_p);
            zin = agg;
        }
        // z (pre-norm) is in buf0
        if (g == 2) {
            l2norm_kernel<<<rowGrid, 256, 0, stream>>>(buf0, out /* z3 */, N_NODES);
        } else {
            l2norm_kernel<<<rowGrid, 256, 0, stream>>>(buf0, buf0, N_NODES);
            // inst head: elu(z@iw1+ib1) @ iw2 + ib2
            gemm128_kernel<<<gemmGrid, 256, gemmLds, stream>>>(buf0, wti1, ib1, buf1, N_NODES, 2, a_p);
            gemm128_kernel<<<gemmGrid, 256, gemmLds, stream>>>(buf1, wti2, ib2,
                out + (size_t)(1 + g) * NH, N_NODES, 0, a_p);
            // cluster head: l2norm(softmax(elu(z@cw1+cb1) @ cw2 + cb2))
            gemm128_kernel<<<gemmGrid, 256, gemmLds, stream>>>(buf0, wtc1, cb1, buf1, N_NODES, 2, a_p);
            cluster_kernel<<<rowGrid, 256, 0, stream>>>(buf1, cw2, cb2,
                out + 3 * NH + (size_t)g * N_NODES * 3, N_NODES);
        }
    }
}